// DCT_1872605741914
// MI455X (gfx1250) — compile-verified
//
#include <hip/hip_runtime.h>

// ---------------------------------------------------------------------------
// DCT-based 1.25x resample:  out = R @ x @ R^T,  R = I960[:, :768] @ D768
// R is precomputed once into workspace, then applied as two fused WMMA GEMMs
// (fp32 WMMA 16x16x4 — data is fp32 and the problem is HBM-bound, so fp32
// matrix ops are free; no precision downgrade needed).
// ---------------------------------------------------------------------------

typedef float v2f __attribute__((ext_vector_type(2)));
typedef float v8f __attribute__((ext_vector_type(8)));

#define H_IN   768
#define W_IN   768
#define H_OUT  960
#define W_OUT  960
#define NIMG   48
#define LDS_STRIDE 772          // 768 + 4 pad -> stride%64banks = 4, conflict-free
#define MT_STRIDE  1024         // padded cols so phase-2 loads are never OOB

// workspace layout (float offsets)
#define OFF_I   ((size_t)0)                              // I960 (weighted): 960x768
#define OFF_D   (OFF_I + (size_t)H_OUT * H_IN)           // D768:            768x768
#define OFF_M   (OFF_D + (size_t)H_IN * H_IN)            // M = I@D:         960x768
#define OFF_MT  (OFF_M + (size_t)H_OUT * H_IN)           // M^T (padded):    768x1024

#define PI_D 3.14159265358979323846

// --- 1) cosine tables with integer-mod argument reduction (args < 2*pi) -----
__global__ void fill_tables(float* __restrict__ ws) {
    int idx = blockIdx.x * blockDim.x + threadIdx.x;
    const int nI = H_OUT * H_IN;
    const int nD = H_IN * H_IN;
    if (idx < nI) {
        int m = idx / H_IN, k = idx - m * H_IN;
        int r = ((2 * m + 1) * k) % (4 * H_OUT);
        float ang = (float)r * (float)(PI_D / (2.0 * H_OUT));
        float w = (k == 0) ? 0.5f : 1.0f;
        ws[OFF_I + (size_t)idx] = (1.0f / (float)H_OUT) * w * cosf(ang);
    } else if (idx < nI + nD) {
        int j = idx - nI;
        int k = j / H_IN, n = j - k * H_IN;
        int r = ((2 * n + 1) * k) % (4 * H_IN);
        float ang = (float)r * (float)(PI_D / (2.0 * H_IN));
        ws[OFF_D + (size_t)j] = 2.0f * cosf(ang);
    }
}

// --- 2) M = I @ D  (960x768x768, one dot product per thread; also writes M^T)
__global__ void build_M(float* __restrict__ ws) {
    int idx = blockIdx.x * blockDim.x + threadIdx.x;
    if (idx >= H_OUT * H_IN) return;
    int m = idx / H_IN, n = idx - m * H_IN;
    const float* Ir = ws + OFF_I + (size_t)m * H_IN;     // row m (broadcast in wave)
    const float* Dc = ws + OFF_D + n;                    // col n (coalesced in wave)
    float s = 0.0f;
    #pragma unroll 4
    for (int k = 0; k < H_IN; ++k)
        s = fmaf(Ir[k], Dc[(size_t)k * H_IN], s);
    ws[OFF_M  + (size_t)idx] = s;
    ws[OFF_MT + (size_t)n * MT_STRIDE + m] = s;
}

// --- 3) fused resample: per (16-row block, image):  T = M_blk @ X ; out = T @ M^T
__launch_bounds__(256)
__global__ void dct_resample(const float* __restrict__ x,
                             float* __restrict__ out,
                             const float* __restrict__ ws) {
    __shared__ float ldsM[16 * LDS_STRIDE];   // staged operator rows
    __shared__ float ldsT[16 * LDS_STRIDE];   // intermediate T block

    const int m0   = blockIdx.x * 16;
    const int img  = blockIdx.y;
    const float* M  = ws + OFF_M;
    const float* MT = ws + OFF_MT;
    const float* X  = x + (size_t)img * H_IN * W_IN;

    const int t      = threadIdx.x;
    const int lane   = t & 31;
    const int laneLo = lane & 15;
    const int laneHi = lane >> 4;
    const int wv     = t >> 5;               // 8 wave32s

    // stage M[m0..m0+15][0..767] into LDS (coalesced)
    for (int i = t; i < 16 * H_IN; i += 256) {
        int r = i / H_IN, c = i - r * H_IN;
        ldsM[r * LDS_STRIDE + c] = M[(size_t)(m0 + r) * H_IN + c];
    }
    __syncthreads();

    // ---- Phase 1: T(16x768) = M_blk(16x768) @ X(768x768) -------------------
    {
        v8f acc[6] = {};
        #pragma unroll 1
        for (int h = 0; h < H_IN; h += 4) {
            v2f a = *reinterpret_cast<const v2f*>(
                        &ldsM[laneLo * LDS_STRIDE + h + 2 * laneHi]);
            #pragma unroll
            for (int q = 0; q < 6; ++q) {
                const int col = (wv * 6 + q) * 16 + laneLo;
                const float* bp = X + (size_t)(h + 2 * laneHi) * W_IN + col;
                v2f b; b.x = bp[0]; b.y = bp[W_IN];
                acc[q] = __builtin_amdgcn_wmma_f32_16x16x4_f32(
                             false, a, false, b, (short)0, acc[q], false, false);
            }
        }
        #pragma unroll
        for (int q = 0; q < 6; ++q) {
            const int w0 = (wv * 6 + q) * 16;
            #pragma unroll
            for (int v = 0; v < 8; ++v)
                ldsT[(v + 8 * laneHi) * LDS_STRIDE + w0 + laneLo] = acc[q][v];
        }
    }
    __syncthreads();

    // ---- Phase 2: out(16x960) = T(16x768) @ MT(768x960) --------------------
    {
        v8f acc[8] = {};
        #pragma unroll 1
        for (int w = 0; w < H_IN; w += 4) {
            v2f a = *reinterpret_cast<const v2f*>(
                        &ldsT[laneLo * LDS_STRIDE + w + 2 * laneHi]);
            #pragma unroll
            for (int q = 0; q < 8; ++q) {
                const int j0 = (wv + 8 * q) * 16;   // cols 960..1023 stay in-bounds
                const float* bp = MT + (size_t)(w + 2 * laneHi) * MT_STRIDE
                                     + j0 + laneLo;
                v2f b; b.x = bp[0]; b.y = bp[MT_STRIDE];
                acc[q] = __builtin_amdgcn_wmma_f32_16x16x4_f32(
                             false, a, false, b, (short)0, acc[q], false, false);
            }
        }
        float* O = out + (size_t)img * H_OUT * W_OUT;
        #pragma unroll
        for (int q = 0; q < 8; ++q) {
            const int jt = wv + 8 * q;
            if (jt < W_OUT / 16) {
                const int j = jt * 16 + laneLo;
                #pragma unroll
                for (int v = 0; v < 8; ++v)
                    O[(size_t)(m0 + v + 8 * laneHi) * W_OUT + j] = acc[q][v];
            }
        }
    }
}

// ---------------------------------------------------------------------------
extern "C" void kernel_launch(void* const* d_in, const int* in_sizes, int n_in,
                              void* d_out, int out_size, void* d_ws, size_t ws_size,
                              hipStream_t stream) {
    const float* x = (const float*)d_in[0];     // (16,3,768,768) fp32 = 48 images
    float* out = (float*)d_out;                 // (16,3,960,960) fp32
    float* ws  = (float*)d_ws;                  // needs ~11.4 MB

    {
        int n = H_OUT * H_IN + H_IN * H_IN;
        fill_tables<<<(n + 255) / 256, 256, 0, stream>>>(ws);
    }
    {
        int n = H_OUT * H_IN;
        build_M<<<(n + 255) / 256, 256, 0, stream>>>(ws);
    }
    {
        dim3 grid(H_OUT / 16, NIMG);
        dct_resample<<<grid, 256, 0, stream>>>(x, out, ws);
    }
}